// TrendAttention_47132971107265
// MI455X (gfx1250) — compile-verified
//
#include <hip/hip_runtime.h>
#include <math.h>

typedef __attribute__((ext_vector_type(2))) float v2f;
typedef __attribute__((ext_vector_type(4))) float v4f;
typedef __attribute__((ext_vector_type(8))) float v8f;

#define NB 8
#define NH 8
#define NS 1024
#define NE 64

// per-iteration tables (stride = it+1)
// Li        = {1024, 512, 342, 256}
// strips    = {  64,  32,  22,  16}   (ceil(Li/16))
// wg bases  = {0, 4096, 6144, 7552}, total 8576
// attn base offsets (floats) inside d_out, after out (4,194,304 floats)

__global__ __launch_bounds__(256)
void trend_attn_qk_kernel(const float* __restrict__ q,
                          const float* __restrict__ k,
                          float* __restrict__ dout,
                          float* __restrict__ colsum)
{
    const int Li[4]        = {1024, 512, 342, 256};
    const int strips[4]    = {64, 32, 22, 16};
    const int wgBase[4]    = {0, 4096, 6144, 7552};
    const size_t attnBase[4] = {4194304ull, 71303168ull, 104857600ull, 127270912ull};

    const int wg = blockIdx.x;
    int it;
    if (wg < 4096)      it = 0;
    else if (wg < 6144) it = 1;
    else if (wg < 7552) it = 2;
    else                it = 3;
    const int local  = wg - wgBase[it];
    const int nst    = strips[it];
    const int bh     = local / nst;
    const int strip  = local - bh * nst;
    const int b      = bh >> 3;
    const int h      = bh & 7;
    const int Lq     = Li[it];
    const int stride = it + 1;

    extern __shared__ float smem[];
    float* attnS = smem;                 // [16][1028] padded rows
    float* rowM  = smem + 16 * 1028;     // [16]
    float* rowZ  = rowM + 16;            // [16]

    const int tid   = threadIdx.x;
    const int lane  = tid & 31;
    const int w     = tid >> 5;          // wave id 0..7
    const int rlane = lane & 15;
    const int hi    = lane >> 4;

    // ---------------- Phase A: WMMA f32 16x16x4, 16x1024 strip ----------------
    const int lq   = strip * 16 + rlane;            // row of strided-Q
    const int qrow = (lq < Lq) ? lq * stride : 0;   // clamp; masked later
    const float* qp = q + (((size_t)b * NS + qrow) * NH + h) * NE;
    v2f afrag[16];
#pragma unroll
    for (int e0 = 0; e0 < 16; ++e0)
        afrag[e0] = *(const v2f*)(qp + e0 * 4 + 2 * hi);

    for (int t = 0; t < 8; ++t) {
        const int sbase = w * 128 + t * 16;
        const float* kp = k + (((size_t)b * NS + (sbase + rlane)) * NH + h) * NE;
        v2f bfrag[16];
#pragma unroll
        for (int e0 = 0; e0 < 16; ++e0)
            bfrag[e0] = *(const v2f*)(kp + e0 * 4 + 2 * hi);
        v8f acc = {0.f, 0.f, 0.f, 0.f, 0.f, 0.f, 0.f, 0.f};
#pragma unroll
        for (int e0 = 0; e0 < 16; ++e0)
            acc = __builtin_amdgcn_wmma_f32_16x16x4_f32(
                false, afrag[e0], false, bfrag[e0], (short)0, acc, false, false);
        // C/D layout: VGPR j -> (M = j + 8*hi, N = rlane)
#pragma unroll
        for (int j = 0; j < 8; ++j)
            attnS[(j + 8 * hi) * 1028 + sbase + rlane] = acc[j];
    }
    __syncthreads();

    // ---------------- Phase B: per-row max and exp-sum ----------------
    {
        const int row = tid >> 4;
        const int sub = tid & 15;
        const float* rp = attnS + row * 1028 + sub * 64;
        float m = -3.402823466e38f;
        for (int c = 0; c < 64; ++c) m = fmaxf(m, rp[c]);
        for (int off = 8; off >= 1; off >>= 1) m = fmaxf(m, __shfl_xor(m, off, 16));
        float z = 0.f;
        for (int c = 0; c < 64; ++c) z += __expf(rp[c] - m);
        for (int off = 8; off >= 1; off >>= 1) z += __shfl_xor(z, off, 16);
        if (sub == 0) { rowM[row] = m; rowZ[row] = z; }
    }
    __syncthreads();

    // ---- Phase C: stream raw logits out (NT) + softmax column-sum atomics ----
    {
        const int c0 = tid * 4;
        float cs0 = 0.f, cs1 = 0.f, cs2 = 0.f, cs3 = 0.f;
        float* attnOut = dout + attnBase[it];
        for (int r = 0; r < 16; ++r) {
            const int lg = strip * 16 + r;
            if (lg >= Lq) break;                      // only last strip of it==2
            const v4f av = *(const v4f*)(attnS + r * 1028 + c0);
            __builtin_nontemporal_store(
                av, (v4f*)(attnOut + (((size_t)bh * Lq + lg) * NS + c0)));
            const float m = rowM[r];
            const float iz = 1.0f / rowZ[r];
            cs0 += __expf(av.x - m) * iz;
            cs1 += __expf(av.y - m) * iz;
            cs2 += __expf(av.z - m) * iz;
            cs3 += __expf(av.w - m) * iz;
        }
        float* cp = colsum + (((size_t)it * 64 + bh) << 10) + c0;
        atomicAdd(cp + 0, cs0);
        atomicAdd(cp + 1, cs1);
        atomicAdd(cp + 2, cs2);
        atomicAdd(cp + 3, cs3);
    }
}

// out[b,s,h,e] = values[b,s,h,e] * 0.25 * sum_i colsum[i,b,h,s]
__global__ __launch_bounds__(256)
void trend_attn_out_kernel(const float* __restrict__ v,
                           const float* __restrict__ colsum,
                           float* __restrict__ dout)
{
    const int gid = blockIdx.x * 256 + threadIdx.x;
    const int idx = gid * 4;                  // base of 4 consecutive e's
    const int h   = (idx >> 6) & 7;
    const int bs  = idx >> 9;                 // b*1024 + s
    const int s   = bs & 1023;
    const int b   = bs >> 10;
    const int bh  = b * 8 + h;
    float f = 0.f;
#pragma unroll
    for (int i = 0; i < 4; ++i)
        f += colsum[(((size_t)i * 64 + bh) << 10) + s];
    f *= 0.25f;
    v4f vv = *(const v4f*)(v + idx);
    vv *= f;
    __builtin_nontemporal_store(vv, (v4f*)(dout + idx));
}

extern "C" void kernel_launch(void* const* d_in, const int* in_sizes, int n_in,
                              void* d_out, int out_size, void* d_ws, size_t ws_size,
                              hipStream_t stream) {
    const float* q = (const float*)d_in[0];
    const float* k = (const float*)d_in[1];
    const float* v = (const float*)d_in[2];
    float* out    = (float*)d_out;
    float* colsum = (float*)d_ws;

    // zero the column-sum accumulator (4*64*1024 floats = 1 MB), capture-safe
    (void)hipMemsetAsync(d_ws, 0, (size_t)4 * 64 * 1024 * sizeof(float), stream);

    const size_t shmem = (16 * 1028 + 32) * sizeof(float);  // ~65.9 KB
    hipLaunchKernelGGL(trend_attn_qk_kernel, dim3(8576), dim3(256), shmem, stream,
                       q, k, out, colsum);
    hipLaunchKernelGGL(trend_attn_out_kernel, dim3(4096), dim3(256), 0, stream,
                       v, colsum, out);
}